// ClassMamba_83665962926844
// MI455X (gfx1250) — compile-verified
//
#include <hip/hip_runtime.h>
#include <hip/hip_bf16.h>
#include <math.h>

// ---------------- problem constants (from reference) ----------------
#define DIMX     512
#define D_INNER  1024
#define D_STATE  16
#define DT_RANK  32
#define B_SZ     4
#define N_TOK    4097
#define RPAD     16          // batch rows (4) padded to one WMMA M-tile

typedef __attribute__((ext_vector_type(2))) float v2f;
typedef __attribute__((ext_vector_type(4))) float f4;
typedef __attribute__((ext_vector_type(8))) float v8f;

__device__ __forceinline__ float silu_f(float v) { return v / (1.0f + __expf(-v)); }
__device__ __forceinline__ float softplus_f(float v) {
    return v > 20.0f ? v : log1pf(__expf(v));
}

// ---------------- 16-row fp32 WMMA GEMM stage, A and C in LDS, W in global:
//   C[16, ntiles*16] = A[16,K] * W[N,K]^T
// one wave per 16x16 tile, waves stride over tiles; EXEC uniform around WMMA.
__device__ __forceinline__ void gemm16_lds(const float* sA, int lda,
                                           const float* __restrict__ W, int ldw,
                                           float* sC, int ldc, int K, int ntiles,
                                           int wv, int lane) {
    const int rm = lane & 15;          // A row (M) / W row (N) within tile
    const int kf = (lane >> 4) << 1;   // lanes 0-15 -> k,k+1 ; lanes 16-31 -> k+2,k+3
    const int mb = (lane >> 4) << 3;   // C: VGPR j holds M=j / M=j+8
    for (int t = wv; t < ntiles; t += 32) {
        const int n0 = t << 4;
        const float* ap = sA + rm * lda + kf;                  // LDS (ds_load_b64)
        const float* wp = W + (long)(n0 + rm) * ldw + kf;      // global
        v8f acc = {0.f, 0.f, 0.f, 0.f, 0.f, 0.f, 0.f, 0.f};
        #pragma unroll 4
        for (int k = 0; k < K; k += 4) {
            v2f a = *(const v2f*)(ap + k);
            v2f b = *(const v2f*)(wp + k);
            acc = __builtin_amdgcn_wmma_f32_16x16x4_f32(
                false, a, false, b, (short)0, acc, false, false);
        }
        #pragma unroll
        for (int j = 0; j < 8; ++j)
            sC[(mb + j) * ldc + n0 + rm] = acc[j];
    }
}

// ---------------- fused cls-token chain: one workgroup, all stages in LDS ----
// LDS map (floats):
//   [0      .. 32767]  s_xz  (16x2048)          ; later: m = [0..8191], u = [8192..16383]
//   [32768  .. 49151]  s_xs  (16x1024)          ; yg computed in place
//   [49152  .. 65535]  s_dq  : cls=[..+8191], q=[+8192..] ; later dt_raw (16x1024)
//   [65536  .. 66559]  s_xdbl (16x64)
// total 66560 floats = 266240 bytes  (< 320KB WGP LDS)
__global__ __launch_bounds__(1024, 1)
void k_cls_chain(const float* __restrict__ x,
                 const float* __restrict__ q_w, const float* __restrict__ in_proj_w,
                 const float* __restrict__ conv_w, const float* __restrict__ conv_b,
                 const float* __restrict__ x_proj_w, const float* __restrict__ dt_w,
                 const float* __restrict__ dt_b, const float* __restrict__ Dp,
                 const float* __restrict__ out_proj_w, const float* __restrict__ proj_w,
                 const float* __restrict__ proj_b, float* __restrict__ out) {
    extern __shared__ float smem[];
    float* s_xz   = smem;            // 32768
    float* s_xs   = smem + 32768;    // 16384
    float* s_dq   = smem + 49152;    // 16384
    float* s_xdbl = smem + 65536;    // 1024
    float* s_cls  = s_dq;
    float* s_q    = s_dq + 8192;

    const int tid  = threadIdx.x;
    const int lane = tid & 31;
    const int wv   = tid >> 5;

    // stage 0: async global->LDS gather of the 4 cls rows (8KB = 512 b128 chunks),
    // issued by 4 full waves (EXEC all-1s); zero-pad rows 4..15.
    if (tid < 128) {
        unsigned long long gbase = (unsigned long long)x;
        unsigned cls_base = (unsigned)(unsigned long long)(void*)s_cls; // LDS byte offset
        #pragma unroll
        for (int i = 0; i < 4; ++i) {
            int chunk = tid + i * 128;                        // 0..511
            int r = chunk >> 7;                               // batch 0..3
            unsigned voff = (unsigned)((r * (N_TOK * DIMX) + (chunk & 127) * 4) * 4);
            unsigned ldsa = cls_base + (unsigned)chunk * 16u;
            asm volatile("global_load_async_to_lds_b128 %0, %1, %2"
                         :: "v"(ldsa), "v"(voff), "s"(gbase) : "memory");
        }
        asm volatile("s_wait_asynccnt 0x0" ::: "memory");
    }
    for (int j = tid; j < (RPAD - B_SZ) * DIMX; j += 1024)
        s_cls[B_SZ * DIMX + j] = 0.0f;
    __syncthreads();

    // stage 1: q = cls @ q_w^T                 (K=512, 32 tiles)
    gemm16_lds(s_cls, DIMX, q_w, DIMX, s_q, DIMX, DIMX, 32, wv, lane);
    __syncthreads();

    // stage 2: xz0 = q @ in_proj_w^T           (K=512, 128 tiles)
    gemm16_lds(s_q, DIMX, in_proj_w, DIMX, s_xz, 2 * D_INNER, DIMX, 128, wv, lane);
    __syncthreads();

    // stage 3: causal conv at t=0 collapses to conv_w[:,3]*xi0 + conv_b ; silu
    for (int g = tid; g < RPAD * D_INNER; g += 1024) {
        int r = g >> 10, d = g & 1023;
        float v = s_xz[r * 2 * D_INNER + d] * conv_w[d * 4 + 3] + conv_b[d];
        s_xs[g] = silu_f(v);
    }
    __syncthreads();

    // stage 4: x_dbl0 = xs0 @ x_proj_w^T       (K=1024, 4 tiles)
    gemm16_lds(s_xs, D_INNER, x_proj_w, D_INNER, s_xdbl, 64, D_INNER, 4, wv, lane);
    __syncthreads();

    // stage 5: dt_raw = x_dbl0[:, :32] @ dt_w^T (K=32, 64 tiles); lda=64 picks dt cols
    gemm16_lds(s_xdbl, 64, dt_w, DT_RANK, s_dq, D_INNER, DT_RANK, 64, wv, lane);
    __syncthreads();

    // stage 6+7: t=0 scan collapse (h0==0 -> A_log dead) + gate, in place in s_xs:
    //   yg = (softplus(dt_raw + dt_b) * u * (B0.C0) + u*D) * silu(z0)
    for (int g = tid; g < RPAD * D_INNER; g += 1024) {
        int r = g >> 10, d = g & 1023;
        const float* roww = s_xdbl + r * (DT_RANK + 2 * D_STATE);
        float s = 0.0f;
        #pragma unroll
        for (int n = 0; n < D_STATE; ++n)
            s += roww[DT_RANK + n] * roww[DT_RANK + D_STATE + n];
        float dlt = softplus_f(s_dq[g] + dt_b[d]);
        float u = s_xs[g];
        float y = dlt * u * s + u * Dp[d];
        float z = s_xz[r * 2 * D_INNER + D_INNER + d];
        s_xs[g] = y * silu_f(z);
    }
    __syncthreads();

    // stage 8: mixed0 = yg @ out_proj_w^T      (K=1024, 32 tiles) -> s_xz[0:8192]
    gemm16_lds(s_xs, D_INNER, out_proj_w, D_INNER, s_xz, DIMX, D_INNER, 32, wv, lane);
    __syncthreads();

    // stage 9: upd_cls = mixed0 @ proj_w^T     (K=512, 32 tiles) -> s_xz[8192:16384]
    gemm16_lds(s_xz, DIMX, proj_w, DIMX, s_xz + 8192, DIMX, DIMX, 32, wv, lane);
    __syncthreads();

    // stage 10: write cls rows (+ proj bias) into d_out
    for (int g = tid; g < B_SZ * DIMX; g += 1024) {
        int b = g >> 9, c = g & 511;
        out[(long)b * N_TOK * DIMX + c] = s_xz[8192 + g] + proj_b[c];
    }
}

// ---------------- bandwidth stage: passthrough of patch rows (33.5MB), float4
__global__ void k_copy_patch(const float* __restrict__ x, float* __restrict__ out) {
    long g = (long)blockIdx.x * blockDim.x + threadIdx.x;  // 4 * 2^19 float4's
    int  b = (int)(g >> 19);
    long i = g & 524287;
    long off = (long)b * N_TOK * (DIMX / 4) + (DIMX / 4) + i;  // 16B aligned
    const f4* src = (const f4*)x;
    f4* dst = (f4*)out;
    __builtin_prefetch(src + off + 4096, 0, 1);   // global_prefetch_b8
    dst[off] = src[off];
}

extern "C" void kernel_launch(void* const* d_in, const int* in_sizes, int n_in,
                              void* d_out, int out_size, void* d_ws, size_t ws_size,
                              hipStream_t stream) {
    const float* x          = (const float*)d_in[0];
    const float* q_w        = (const float*)d_in[1];
    const float* in_proj_w  = (const float*)d_in[2];
    const float* conv_w     = (const float*)d_in[3];
    const float* conv_b     = (const float*)d_in[4];
    const float* x_proj_w   = (const float*)d_in[5];
    const float* dt_w       = (const float*)d_in[6];
    const float* dt_b       = (const float*)d_in[7];
    // d_in[8] = A_log : provably dead at t=0 (h0 == 0)
    const float* Dp         = (const float*)d_in[9];
    const float* out_proj_w = (const float*)d_in[10];
    const float* proj_w     = (const float*)d_in[11];
    const float* proj_b     = (const float*)d_in[12];
    float* out = (float*)d_out;

    // launch the 33.5MB passthrough first so it overlaps the cls chain
    k_copy_patch<<<(B_SZ * 524288) / 256, 256, 0, stream>>>(x, out);

    // one fused workgroup: entire cls chain, 260KB dynamic LDS (WGP has 320KB)
    const size_t lds_bytes = 66560u * sizeof(float);   // 266240
    k_cls_chain<<<1, 1024, lds_bytes, stream>>>(x, q_w, in_proj_w, conv_w, conv_b,
                                                x_proj_w, dt_w, dt_b, Dp,
                                                out_proj_w, proj_w, proj_b, out);
}